// LIFBase_29824252903749
// MI455X (gfx1250) — compile-verified
//
#include <hip/hip_runtime.h>
#include <stdint.h>

// LIF spiking RNN, fused persistent kernel for MI455X (gfx1250).
// 64 WGs = 4 batch-groups x 16 column-slices; per-step cross-WG spike exchange
// only inside a 16-WG group (cluster-sized) via bit-packed spikes in L2 +
// monotonic atomic barrier + cluster-barrier hint. fp32 tensor path
// (V_WMMA_F32_16X16X4_F32) throughout; W_rec slice + x slab LDS-resident.

typedef float v2f __attribute__((ext_vector_type(2)));
typedef float v8f __attribute__((ext_vector_type(8)));

#define T_STEPS 512
#define BATCH   64
#define D_INP   512
#define HID     1024
#define KP      1028   // padded K stride (floats) for W_rec slice in LDS
#define SBW     33     // padded word stride for spike-bit rows in LDS
#define XP      516    // padded K stride (floats) for x slab in LDS

// dyn LDS dwords: W_rec slice + spike bits + split-K partials + x slab
#define LDS_DW (64 * KP + 16 * SBW + 4 * 256 + 2 * 4 * 256 + 16 * XP)
#define LDS_BYTES (LDS_DW * 4)

static __device__ __forceinline__ v8f wmma4(v2f a, v2f b, v8f c) {
  return __builtin_amdgcn_wmma_f32_16x16x4_f32(false, a, false, b, (short)0, c, false, false);
}

__launch_bounds__(256, 1)
__global__ void lif_fused(const float* __restrict__ x,
                          const float* __restrict__ Win,
                          const float* __restrict__ Wrec,
                          const float* __restrict__ leak_p,
                          const float* __restrict__ vth_p,
                          float* __restrict__ out,
                          unsigned* __restrict__ gbits,   // [2][64 rows][32 words]
                          unsigned* __restrict__ gcnt)    // [4] barrier counters
{
  extern __shared__ float smem[];
  float*    wrec  = smem;                              // 64 x KP
  unsigned* sbits = (unsigned*)(wrec + 64 * KP);       // 16 x SBW
  float*    recB  = (float*)(sbits + 16 * SBW);        // 4 tiles x 256
  float*    inpB  = recB + 4 * 256;                    // 2 parities x 4 tiles x 256
  float*    xs    = inpB + 2 * 4 * 256;                // 16 x XP (x slab)

  const int tid   = threadIdx.x;
  const int wave  = tid >> 5;
  const int lane  = tid & 31;
  const int bid   = blockIdx.x;      // 0..63
  const int grp   = bid >> 4;        // batch group (16 rows)
  const int sl    = bid & 15;        // 64-column slice
  const int bb    = grp * 16;
  const int ncol0 = sl * 64;
  const bool front = (wave < 4);
  const int tile  = wave & 3;        // 16x16 output tile within slice
  const int n0    = ncol0 + tile * 16;
  const int mlo   = lane & 15;
  const int koff  = (lane >> 4) << 1;  // 0 or 2 (fragment K sub-offset)
  const int kbase = front ? 0 : 256;   // split-K half for this wave

  const float leak = leak_p[0];
  const float vth  = vth_p[0];
  const float rst  = vth * leak;

  // hoisted fragment base pointers (constant K trip counts keep loops uniform)
  const float* wb_frag = Win + (size_t)(n0 + mlo) * D_INP + kbase + koff;
  const float* xs_frag = xs + mlo * XP + kbase + koff;

  // ---- stage W_rec slice (rows ncol0..+63, all K) into LDS ----
  {
    const int nl = tid >> 2;
    const int q  = tid & 3;
    const float4* src = (const float4*)(Wrec + (size_t)(ncol0 + nl) * HID + q * 256);
    float4* dst = (float4*)(wrec + nl * KP + q * 256);
    #pragma unroll 4
    for (int c = 0; c < 64; ++c) dst[c] = src[c];
  }
  // ---- stage x slab for t=0 ----
  {
    const int r  = tid >> 4;
    const int c4 = tid & 15;
    const float4* src = (const float4*)(x + (size_t)(bb + r) * D_INP) + c4;
    float4* dst = (float4*)(xs + r * XP) + c4;
    #pragma unroll
    for (int u = 0; u < 8; ++u) dst[u * 16] = src[u * 16];
  }
  for (int idx = tid; idx < 16 * SBW; idx += 256) sbits[idx] = 0u;  // t=0: no spikes
  __syncthreads();

  // ---- input projection partial for t=0 ----
  v8f accI = {};
  #pragma unroll 16
  for (int u = 0; u < 64; ++u) {
    v2f a = *(const v2f*)(xs_frag + u * 4);
    v2f b = *(const v2f*)(wb_frag + u * 4);
    accI = wmma4(a, b, accI);
  }
  if (!front) {
    float* d = inpB + tile * 256;  // parity 0
    #pragma unroll
    for (int j = 0; j < 8; ++j) d[j * 32 + lane] = accI[j];
  }
  __syncthreads();

  v8f hreg  = {};
  v8f sprev = {};
  float* out_s = out;
  float* out_h = out + (size_t)T_STEPS * BATCH * HID;

  for (int t = 0; t < T_STEPS; ++t) {
    const int pr = t & 1;        // parity read (spikes feeding step t)
    const int pw = pr ^ 1;       // parity written (spikes produced at step t)

    // stage x slab for t+1 (slab for t was consumed at end of step t-1)
    if (t + 1 < T_STEPS) {
      const int r  = tid >> 4;
      const int c4 = tid & 15;
      const float4* src =
          (const float4*)(x + (size_t)((t + 1) * BATCH + bb + r) * D_INP) + c4;
      float4* dst = (float4*)(xs + r * XP) + c4;
      #pragma unroll
      for (int u = 0; u < 8; ++u) dst[u * 16] = src[u * 16];
    }
    if (t > 0) {  // pull this group's 16 rows of packed spikes (2KB) from L2
      const int m  = tid >> 4;
      const int wi = (tid & 15) << 1;
      const unsigned* src = gbits + pr * 2048 + (bb + m) * 32 + wi;
      sbits[m * SBW + wi]     = src[0];
      sbits[m * SBW + wi + 1] = src[1];
      __syncthreads();
    }

    // ---- recurrent GEMM partial: s_prev(bits) @ W_rec^T over this wave's K half ----
    v8f accR = {};
    {
      const int w0 = front ? 0 : 16;
      const int nl = tile * 16 + mlo;
      const float* brow0 = wrec + nl * KP + koff + w0 * 32;
      #pragma unroll 2
      for (int u = 0; u < 16; ++u) {
        const unsigned bits = sbits[mlo * SBW + w0 + u];
        const float* brow = brow0 + u * 32;
        #pragma unroll
        for (int j = 0; j < 8; ++j) {
          const int sh = (j << 2) + koff;
          v2f a;
          a.x = ((bits >> sh) & 1u) ? 1.0f : 0.0f;
          a.y = ((bits >> (sh + 1)) & 1u) ? 1.0f : 0.0f;
          v2f b = *(const v2f*)(brow + (j << 2));
          accR = wmma4(a, b, accR);
        }
      }
    }
    if (!front) {
      float* d = recB + tile * 256;
      #pragma unroll
      for (int j = 0; j < 8; ++j) d[j * 32 + lane] = accR[j];
    }
    __syncthreads();  // also orders xs slab writes before inp-partial reads below

    if (front) {
      const float* rB = recB + tile * 256;
      const float* iB = inpB + pr * 1024 + tile * 256;
      unsigned short* g16 = (unsigned short*)(gbits + pw * 2048);
      const int s16 = (sl << 2) + tile;
      const size_t obase = (size_t)t * BATCH * HID + (size_t)bb * HID + n0 + mlo;
      #pragma unroll
      for (int j = 0; j < 8; ++j) {
        float h = accI[j] + iB[j * 32 + lane] + accR[j] + rB[j * 32 + lane]
                + leak * hreg[j] - sprev[j] * rst;
        float sj = (h >= vth) ? 1.0f : 0.0f;
        const int mrow = j + ((lane >> 4) << 3);
        const size_t idx = obase + (size_t)mrow * HID;
        out_h[idx] = h;
        out_s[idx] = sj;
        const unsigned mk = __builtin_amdgcn_ballot_w32(sj != 0.0f);
        if (lane == 0) g16[(bb + j) * 64 + s16]     = (unsigned short)(mk & 0xffffu);
        if (lane == 1) g16[(bb + j + 8) * 64 + s16] = (unsigned short)(mk >> 16);
        hreg[j]  = h;
        sprev[j] = sj;
      }
    }

    if (t + 1 < T_STEPS) {
      // overlap: input projection partial for t+1 (independent of spikes)
      {
        v8f c = {};
        #pragma unroll 16
        for (int u = 0; u < 64; ++u) {
          v2f a = *(const v2f*)(xs_frag + u * 4);
          v2f b = *(const v2f*)(wb_frag + u * 4);
          c = wmma4(a, b, c);
        }
        accI = c;
      }
      if (!front) {
        float* d = inpB + pw * 1024 + tile * 256;
        #pragma unroll
        for (int j = 0; j < 8; ++j) d[j * 32 + lane] = accI[j];
      }
      if (t + 2 < T_STEPS)  // warm L2 for the next slab stage
        __builtin_prefetch(x + (size_t)((t + 2) * BATCH + bb + mlo) * D_INP, 0, 1);

      // ---- inter-WG sync among the 16 WGs of this batch-group ----
      __builtin_amdgcn_fence(__ATOMIC_RELEASE, "agent");  // publish spike stores
      __syncthreads();
      if (tid == 0) {
        __hip_atomic_fetch_add(&gcnt[grp], 1u, __ATOMIC_RELEASE, __HIP_MEMORY_SCOPE_AGENT);
        const unsigned tgt = 16u * (unsigned)(t + 1);     // monotonic, no reset
        while (__hip_atomic_load(&gcnt[grp], __ATOMIC_ACQUIRE, __HIP_MEMORY_SCOPE_AGENT) < tgt)
          __builtin_amdgcn_s_sleep(1);
      }
      if (wave == 0)  // cluster-barrier hint: NOP if not cluster-dispatched
        asm volatile("s_barrier_signal -3\n\ts_barrier_wait -3" ::: "memory");
      __syncthreads();
      __builtin_amdgcn_fence(__ATOMIC_ACQUIRE, "agent");  // observe other WGPs' spikes
    }
  }
}

extern "C" void kernel_launch(void* const* d_in, const int* in_sizes, int n_in,
                              void* d_out, int out_size, void* d_ws, size_t ws_size,
                              hipStream_t stream) {
  (void)in_sizes; (void)n_in; (void)out_size; (void)ws_size;
  const float* x    = (const float*)d_in[0];
  const float* Win  = (const float*)d_in[1];
  const float* Wrec = (const float*)d_in[2];
  const float* leak = (const float*)d_in[3];
  const float* vth  = (const float*)d_in[4];
  float* out = (float*)d_out;

  unsigned* gbits = (unsigned*)d_ws;        // 2 * 64 * 32 u32 = 16KB
  unsigned* gcnt  = gbits + 2 * 2048;       // 4 counters

  hipMemsetAsync(d_ws, 0, 2 * 2048 * 4 + 64, stream);  // capture-safe init

  hipFuncSetAttribute(reinterpret_cast<const void*>(lif_fused),
                      hipFuncAttributeMaxDynamicSharedMemorySize, LDS_BYTES);

  lif_fused<<<dim3(64), dim3(256), LDS_BYTES, stream>>>(
      x, Win, Wrec, leak, vth, out, gbits, gcnt);
}